// GCNLayer_1219770712715
// MI455X (gfx1250) — compile-verified
//
#include <hip/hip_runtime.h>
#include <hip/hip_bf16.h>

typedef float v2f __attribute__((ext_vector_type(2)));
typedef float v8f __attribute__((ext_vector_type(8)));

#define D_FEAT 128

// ---------------- degree / normalization ----------------

__global__ void init_deg_kernel(float* __restrict__ deg, int n) {
    int i = blockIdx.x * blockDim.x + threadIdx.x;
    if (i < n) deg[i] = 1.0f;  // self loop contributes 1 to every node's degree
}

__global__ void count_deg_kernel(const int* __restrict__ rows, float* __restrict__ deg, int e) {
    int i = blockIdx.x * blockDim.x + threadIdx.x;
    if (i < e) atomicAdd(&deg[rows[i]], 1.0f);
}

__global__ void rsqrt_kernel(float* __restrict__ deg, int n) {
    int i = blockIdx.x * blockDim.x + threadIdx.x;
    if (i < n) deg[i] = rsqrtf(deg[i]);
}

// ---------------- dense projection: h = x @ W^T via fp32 WMMA ----------------
// One wave computes a 16x16 tile of h. A = x rows (16x4 slice per step),
// B[k][n] = W[n][k] (W is [out,in] row-major, so h = x @ W^T).
__global__ void __launch_bounds__(256) gemm_h_kernel(const float* __restrict__ x,
                                                     const float* __restrict__ W,
                                                     float* __restrict__ h,
                                                     int n_nodes) {
    const int wave  = blockIdx.x * (blockDim.x >> 5) + (threadIdx.x >> 5);
    const int lane  = threadIdx.x & 31;
    const int tiles_n = D_FEAT / 16;              // 8 column tiles
    const int m0 = (wave / tiles_n) * 16;         // output row tile
    const int n0 = (wave % tiles_n) * 16;         // output col tile
    if (m0 >= n_nodes) return;                    // wave-uniform -> EXEC all-1 at WMMA

    const int half = lane >> 4;                   // 0: K,K+1  1: K+2,K+3 (ISA A layout)
    const int l15  = lane & 15;
    const size_t baseA = (size_t)(m0 + l15) * D_FEAT + (size_t)(half * 2);
    const size_t baseB = (size_t)(n0 + l15) * D_FEAT + (size_t)(half * 2);

    v8f c = {};
#pragma unroll
    for (int k = 0; k < D_FEAT; k += 4) {
        v2f a = *(const v2f*)(x + baseA + k);
        v2f b = *(const v2f*)(W + baseB + k);
        // 8 args: (neg_a, A, neg_b, B, c_mod, C, reuse_a, reuse_b)
        c = __builtin_amdgcn_wmma_f32_16x16x4_f32(false, a, false, b,
                                                  (short)0, c, false, false);
    }

    // D layout: VGPR v holds row m0+v (lanes 0-15) / m0+8+v (lanes 16-31), col n0+l15
    float* outp = h + (size_t)(m0 + half * 8) * D_FEAT + n0 + l15;
#pragma unroll
    for (int v = 0; v < 8; ++v) outp[(size_t)v * D_FEAT] = c[v];
}

// ---------------- self-loop term: out = (dr[i]^2) * h[i,:] ----------------
__global__ void self_loop_kernel(const float* __restrict__ h,
                                 const float* __restrict__ dr,
                                 float* __restrict__ out, long total) {
    long i = (long)blockIdx.x * blockDim.x + threadIdx.x;
    if (i < total) {
        int node = (int)(i >> 7);                 // /128
        float s = dr[node];
        out[i] = s * s * h[i];
    }
}

// ---------------- edge aggregation: one wave32 per edge ----------------
// lane L handles floats [4L, 4L+4): float4 gather from h[col], 4 f32 atomics
// into out[row]. h (51 MB) is resident in the 192 MB L2, so gathers + atomics
// are L2-local on MI455X.
__global__ void __launch_bounds__(256) edge_agg_kernel(const int* __restrict__ rows,
                                                       const int* __restrict__ cols,
                                                       const float* __restrict__ h,
                                                       const float* __restrict__ dr,
                                                       float* __restrict__ out, int e) {
    long gid  = (long)blockIdx.x * blockDim.x + threadIdx.x;
    int  edge = (int)(gid >> 5);
    int  lane = (int)(gid & 31);
    if (edge >= e) return;
    int r = rows[edge];
    int c = cols[edge];
    float v = dr[r] * dr[c];
    const float4* hc = (const float4*)(h + (size_t)c * D_FEAT);
    float4 hv = hc[lane];
    float* o = out + (size_t)r * D_FEAT + lane * 4;
    atomicAdd(o + 0, v * hv.x);
    atomicAdd(o + 1, v * hv.y);
    atomicAdd(o + 2, v * hv.z);
    atomicAdd(o + 3, v * hv.w);
}

// ---------------- ReLU ----------------
__global__ void relu_kernel(float* __restrict__ out, long total) {
    long i = (long)blockIdx.x * blockDim.x + threadIdx.x;
    if (i < total) out[i] = fmaxf(out[i], 0.0f);
}

extern "C" void kernel_launch(void* const* d_in, const int* in_sizes, int n_in,
                              void* d_out, int out_size, void* d_ws, size_t ws_size,
                              hipStream_t stream) {
    (void)n_in; (void)out_size; (void)ws_size;

    const float* x  = (const float*)d_in[0];
    const int*   ei = (const int*)d_in[1];      // [2, E] row-major
    const float* W  = (const float*)d_in[2];    // [D_OUT, D_IN] row-major
    float* out = (float*)d_out;

    const int n_nodes = in_sizes[0] / D_FEAT;   // 100000
    const int e       = in_sizes[1] / 2;        // 1600000
    const int* rows = ei;
    const int* cols = ei + e;

    // workspace layout: dr [n_nodes floats] | h [n_nodes * 128 floats]
    float* dr = (float*)d_ws;
    size_t h_off = (((size_t)n_nodes * sizeof(float)) + 255) & ~(size_t)255;
    float* h = (float*)((char*)d_ws + h_off);

    const long total = (long)n_nodes * D_FEAT;

    init_deg_kernel<<<(n_nodes + 255) / 256, 256, 0, stream>>>(dr, n_nodes);
    count_deg_kernel<<<(e + 255) / 256, 256, 0, stream>>>(rows, dr, e);
    rsqrt_kernel<<<(n_nodes + 255) / 256, 256, 0, stream>>>(dr, n_nodes);

    // 16-row tiles * 8 col-tiles, 8 waves per 256-thread block
    int row_tiles = (n_nodes + 15) / 16;
    long waves = (long)row_tiles * (D_FEAT / 16);
    int gemm_blocks = (int)((waves + 7) / 8);
    gemm_h_kernel<<<gemm_blocks, 256, 0, stream>>>(x, W, h, n_nodes);

    self_loop_kernel<<<(int)((total + 255) / 256), 256, 0, stream>>>(h, dr, out, total);

    long agg_threads = (long)e * 32;
    edge_agg_kernel<<<(int)((agg_threads + 255) / 256), 256, 0, stream>>>(rows, cols, h, dr, out, e);

    relu_kernel<<<(int)((total + 255) / 256), 256, 0, stream>>>(out, total);
}